// HG_block_14826227105923
// MI455X (gfx1250) — compile-verified
//
#include <hip/hip_runtime.h>
#include <math.h>

typedef __attribute__((ext_vector_type(2))) float v2f;
typedef __attribute__((ext_vector_type(8))) float v8f;

#define B_   8
#define C_   192
#define Hh_  56
#define Ww_  56
#define N_   3136     // 56*56
#define NP_  784      // 28*28
#define PH_  28
#define M_   196      // centroids
#define MP_  208      // padded to 13*16
#define C4_  768
#define C2_  384
#define BNSC 0.9999950000374997f   // 1/sqrt(1+1e-5)
#define INVSQC 0.0721687836487032f // 1/sqrt(192)

// ---------------------------------------------------------------------------
// fp32 WMMA GEMM: D[b,m,n] = epilogue( sum_k A[b,m,k]*B[b,k,n] )
// A element at A + b*aB + m*aM + k*aK ; B element at Bm + b*bB + k*BK + n*bN
// BK is a compile-time template parameter so all per-step B offsets fold into
// global_load immediate offsets (one pointer bump per 4 k-steps); BK==1 merges
// the per-lane k-pair into a single b64 load.
// A panel (16 x K) for the block's m-tile is staged in LDS ([k][m] layout,
// bank-conflict-free). One 16x16 output tile per wave, K consumed 4/step with
// V_WMMA_F32_16X16X4_F32, unrolled x4. wave32; EXEC all-ones at every WMMA
// (no early exit; tail waves clamp their tile and skip only the store).
// Out-of-range B columns are CLAMPED (padded dim never appears as K, so
// duplicated values only reach padded outputs; fully deterministic).
// mode: 0 plain, 1 conv-BN affine (+optional exact GELU), 2 sqrt-distance,
//       3 divide-by-colsum.  Optional residual R.
// ---------------------------------------------------------------------------
template <int BK>
__global__ void wmma_gemm_f32(
    const float* __restrict__ A, long long aB, int aM, int aK,
    const float* __restrict__ Bm, long long bB, int bN, int bNvalid,
    float* __restrict__ D, long long dB, int dM,
    int Ntiles, int Ksteps,
    int mode, int gelu,
    const float* __restrict__ wb, const float* __restrict__ wg,
    const float* __restrict__ wbe,
    const float* __restrict__ R, long long rB, int rM,
    const float* __restrict__ aux, long long auxB)
{
    extern __shared__ float ldsA[];          // [k][m] : k*16 + m
    const int wv   = threadIdx.x >> 5;
    const int lane = threadIdx.x & 31;
    const int mt   = blockIdx.x;
    const int nt   = blockIdx.y * 8 + wv;
    const int b    = blockIdx.z;
    const int m0   = mt * 16;

    // ---- cooperative fill of the A panel (16 rows x K) into LDS ----
    const int K     = Ksteps * 4;
    const int total = 16 * K;                // multiple of 256 for all calls
    {
        const float* Abase = A + (long long)b * aB;
        for (int idx = threadIdx.x; idx < total; idx += 256) {
            const int m = idx & 15;
            const int k = idx >> 4;
            ldsA[idx] = Abase[(long long)(m0 + m) * aM + (long long)k * aK];
        }
    }
    __syncthreads();

    // ---- per-wave tile ----
    const int  ntEff = nt < Ntiles ? nt : Ntiles - 1;   // clamp, keep wave alive
    const int  n0    = ntEff * 16;
    const int  half  = lane >> 4;            // 0 -> K, 1 -> K+2 (ISA f32 layout)
    const int  lq    = lane & 15;
    const int  ncol  = n0 + lq;
    const int  nc    = ncol < bNvalid ? ncol : bNvalid - 1;   // clamp B column

    const float* Bp  = Bm + (long long)b * bB + (long long)nc * (long long)bN
                          + (long long)(2 * half) * BK;
    const float* Ap  = ldsA + 2 * half * 16 + lq;

    v8f acc = {0.f, 0.f, 0.f, 0.f, 0.f, 0.f, 0.f, 0.f};

    const int Ks4 = Ksteps >> 2;             // Ksteps is a multiple of 4
    for (int ks4 = 0; ks4 < Ks4; ++ks4) {
        __builtin_prefetch(Bp + 16 * BK, 0, 1);
#pragma unroll
        for (int j = 0; j < 4; ++j) {
            v2f a, bbv;
            a.x   = Ap[j * 64];              // ds immediates
            a.y   = Ap[j * 64 + 16];
            bbv.x = Bp[j * 4 * BK];          // global immediates (BK constexpr)
            bbv.y = Bp[j * 4 * BK + BK];
            acc = __builtin_amdgcn_wmma_f32_16x16x4_f32(
                      false, a, false, bbv, (short)0, acc, false, false);
        }
        Ap += 256;                           // 16 k-values * 16 lanes
        Bp += 16 * BK;
    }

    if (nt >= Ntiles) return;                // uniform per wave, after all WMMAs

    // C/D layout: VGPR r holds row m0+r (lanes 0-15) and m0+r+8 (lanes 16-31)
#pragma unroll
    for (int r = 0; r < 8; ++r) {
        const int m = m0 + r + 8 * half;
        float v = acc[r];
        if (mode == 1) {
            const float s = wg[m] * BNSC;
            v = s * v + (s * wb[m] + wbe[m]);
            if (gelu) v = 0.5f * v * (1.f + erff(v * 0.70710678118654752f));
        } else if (mode == 2) {
            const float d2 = aux[(long long)b * auxB + m]
                           + aux[(long long)b * auxB + ncol] - 2.f * v;
            v = sqrtf(fmaxf(d2, 1e-12f)) * INVSQC;
        } else if (mode == 3) {
            v = v / (aux[(long long)b * auxB + ncol] + 1e-6f);
        }
        if (R) v += R[(long long)b * rB + (long long)m * rM + ncol];
        D[(long long)b * dB + (long long)m * dM + ncol] = v;
    }
}

// host-side templated launcher so BK folds into immediates
template <int BK>
static void launch_gemm_bk(hipStream_t stream,
                           const float* A, long long aB, int aM, int aK,
                           const float* Bm, long long bB, int bN, int bNv,
                           float* D, long long dB, int dM,
                           int Mt, int Nt, int Ks, int mode, int gelu,
                           const float* wb, const float* wg, const float* wbe,
                           const float* R, long long rB, int rM,
                           const float* aux, long long auxB)
{
    dim3 grid(Mt, (Nt + 7) / 8, B_);
    const size_t shmem = (size_t)256 * (size_t)Ks;   // 16 * K * 4 bytes
    wmma_gemm_f32<BK><<<grid, 256, shmem, stream>>>(A, aB, aM, aK, Bm, bB, bN, bNv,
                                                    D, dB, dM, Nt, Ks, mode, gelu,
                                                    wb, wg, wbe, R, rB, rM, aux, auxB);
}

// ---------------------------------------------------------------------------
// 2x2 average pool: y[b,c,j] (c-major) and pts[b,j,c] (point-major)
// ---------------------------------------------------------------------------
__global__ void pool_kernel(const float* __restrict__ xf,
                            float* __restrict__ y, float* __restrict__ pts)
{
    const long long idx = (long long)blockIdx.x * blockDim.x + threadIdx.x;
    if (idx >= (long long)B_ * C_ * NP_) return;
    const int j = (int)(idx % NP_);
    const int t = (int)(idx / NP_);
    const int c = t % C_, b = t / C_;
    const int ph = j / PH_, pw = j % PH_;
    const float* p = xf + ((long long)b * C_ + c) * N_ + (2 * ph) * Ww_ + 2 * pw;
    const float v = 0.25f * (p[0] + p[1] + p[Ww_] + p[Ww_ + 1]);
    y[idx] = v;
    pts[((long long)b * NP_ + j) * C_ + c] = v;
}

__global__ void sq_kernel(const float* __restrict__ pts, float* __restrict__ sq)
{
    const int idx = blockIdx.x * blockDim.x + threadIdx.x;
    if (idx >= B_ * NP_) return;
    const float* p = pts + (long long)idx * C_;
    float s = 0.f;
    for (int c = 0; c < C_; ++c) s += p[c] * p[c];
    sq[idx] = s;
}

__global__ void init_dmax_kernel(int* dmax)
{
    if (threadIdx.x < B_) dmax[threadIdx.x] = 0;
}

// density_i = exp(-mean(d^2 of 5 nearest)), plus per-batch max distance
__global__ void density_kernel(const float* __restrict__ dist,
                               float* __restrict__ density, int* __restrict__ dmax)
{
    const int idx = blockIdx.x * blockDim.x + threadIdx.x;
    if (idx >= B_ * NP_) return;
    const int b = idx / NP_;
    const float* row = dist + (long long)idx * NP_;
    float best[5] = {3e38f, 3e38f, 3e38f, 3e38f, 3e38f};
    float rmax = 0.f;
    for (int j = 0; j < NP_; ++j) {
        const float d = row[j];
        rmax = fmaxf(rmax, d);
        if (d < best[4]) {
            best[4] = d;
#pragma unroll
            for (int t = 4; t > 0; --t)
                if (best[t] < best[t - 1]) { float tt = best[t]; best[t] = best[t - 1]; best[t - 1] = tt; }
        }
    }
    float s = 0.f;
#pragma unroll
    for (int t = 0; t < 5; ++t) s += best[t] * best[t];
    density[idx] = expf(-s * 0.2f);
    atomicMax(&dmax[b], __float_as_int(rmax));   // dist >= 0: int-bits ordered
}

__global__ void score_kernel(const float* __restrict__ dist,
                             const float* __restrict__ density,
                             const int* __restrict__ dmax,
                             float* __restrict__ score)
{
    const int idx = blockIdx.x * blockDim.x + threadIdx.x;
    if (idx >= B_ * NP_) return;
    const int b = idx / NP_;
    const float di = density[idx];
    const float* row = dist + (long long)idx * NP_;
    const float* db  = density + (long long)b * NP_;
    float mn = __int_as_float(dmax[b]);
    for (int j = 0; j < NP_; ++j)
        if (db[j] > di) mn = fminf(mn, row[j]);
    score[idx] = mn * di;
}

// exact stable top-196 by rank (bijective with index tie-break), gathers pts
__global__ void select_kernel(const float* __restrict__ score,
                              const float* __restrict__ pts,
                              float* __restrict__ cent)
{
    const int idx = blockIdx.x * blockDim.x + threadIdx.x;
    if (idx >= B_ * NP_) return;
    const int b = idx / NP_, i = idx % NP_;
    const float s = score[idx];
    const float* sb = score + (long long)b * NP_;
    int r = 0;
    for (int j = 0; j < NP_; ++j) {
        const float sj = sb[j];
        r += (sj > s) || (sj == s && j < i);
    }
    if (r < M_) {
        const float* src = pts + (long long)idx * C_;
        float* dst = cent + ((long long)b * M_ + r) * C_;
        for (int c = 0; c < C_; ++c) dst[c] = src[c];
    }
}

__global__ void softmax_kernel(float* __restrict__ assign)
{
    const long long idx = (long long)blockIdx.x * blockDim.x + threadIdx.x;
    if (idx >= (long long)B_ * N_) return;
    float* row = assign + idx * MP_;
    float mx = -3e38f;
    for (int m = 0; m < M_; ++m) mx = fmaxf(mx, row[m]);
    float s = 0.f;
    for (int m = 0; m < M_; ++m) { const float e = expf(row[m] - mx); row[m] = e; s += e; }
    const float inv = 1.f / s;
    for (int m = 0; m < M_; ++m) row[m] *= inv;
}

__global__ void colsum_kernel(const float* __restrict__ assign, float* __restrict__ colsum)
{
    const int idx = blockIdx.x * blockDim.x + threadIdx.x;
    if (idx >= B_ * MP_) return;
    const int b = idx / MP_, m = idx % MP_;
    const float* base = assign + (long long)b * N_ * MP_ + m;
    float s = 0.f;
    for (int n = 0; n < N_; ++n) s += base[(long long)n * MP_];
    colsum[idx] = s;
}

__global__ void topk5_kernel(const float* __restrict__ assign, int* __restrict__ nn_idx)
{
    const long long idx = (long long)blockIdx.x * blockDim.x + threadIdx.x;
    if (idx >= (long long)B_ * N_) return;
    const float* row = assign + idx * MP_;
    float bv[5] = {-3e38f, -3e38f, -3e38f, -3e38f, -3e38f};
    int   bi[5] = {0, 0, 0, 0, 0};
    for (int m = 0; m < M_; ++m) {
        const float v = row[m];
        if (v > bv[4]) {
            bv[4] = v; bi[4] = m;
#pragma unroll
            for (int t = 4; t > 0; --t)
                if (bv[t] > bv[t - 1]) {
                    float tv = bv[t]; bv[t] = bv[t - 1]; bv[t - 1] = tv;
                    int   ti = bi[t]; bi[t] = bi[t - 1]; bi[t - 1] = ti;
                }
        }
    }
    int* o = nn_idx + idx * 5;
#pragma unroll
    for (int t = 0; t < 5; ++t) o[t] = bi[t];
}

// xcat[b,2c,n]=xf ; xcat[b,2c+1,n]= max_t refined[c, nn_idx[n,t]] - xf
__global__ void xcat_kernel(const float* __restrict__ xf,
                            const float* __restrict__ refined,
                            const int* __restrict__ nn_idx,
                            float* __restrict__ xcat)
{
    const long long idx = (long long)blockIdx.x * blockDim.x + threadIdx.x;
    if (idx >= (long long)B_ * C_ * N_) return;
    const int n = (int)(idx % N_);
    const int t = (int)(idx / N_);
    const int c = t % C_, b = t / C_;
    const int* ni = nn_idx + ((long long)b * N_ + n) * 5;
    const float* rb = refined + ((long long)b * C_ + c) * MP_;
    float mx = -3e38f;
#pragma unroll
    for (int k = 0; k < 5; ++k) mx = fmaxf(mx, rb[ni[k]]);
    const float xv = xf[((long long)b * C_ + c) * N_ + n];
    float* xc = xcat + (long long)b * C2_ * N_;
    xc[(long long)(2 * c) * N_ + n]     = xv;
    xc[(long long)(2 * c + 1) * N_ + n] = mx - xv;
}

// ---------------------------------------------------------------------------
extern "C" void kernel_launch(void* const* d_in, const int* in_sizes, int n_in,
                              void* d_out, int out_size, void* d_ws, size_t ws_size,
                              hipStream_t stream)
{
    // setup_inputs() insertion order: x, then params {fc1, ffn1, ffn2, nn, fc2},
    // each block {w, b, g, be}
    const float* x      = (const float*)d_in[0];
    const float* w_fc1  = (const float*)d_in[1];
    const float* b_fc1  = (const float*)d_in[2];
    const float* g_fc1  = (const float*)d_in[3];
    const float* be_fc1 = (const float*)d_in[4];
    const float* w_f1   = (const float*)d_in[5];
    const float* b_f1   = (const float*)d_in[6];
    const float* g_f1   = (const float*)d_in[7];
    const float* be_f1  = (const float*)d_in[8];
    const float* w_f2   = (const float*)d_in[9];
    const float* b_f2   = (const float*)d_in[10];
    const float* g_f2   = (const float*)d_in[11];
    const float* be_f2  = (const float*)d_in[12];
    const float* w_nn   = (const float*)d_in[13];
    const float* b_nn   = (const float*)d_in[14];
    const float* g_nn   = (const float*)d_in[15];
    const float* be_nn  = (const float*)d_in[16];
    const float* w_fc2  = (const float*)d_in[17];
    const float* b_fc2  = (const float*)d_in[18];
    const float* g_fc2  = (const float*)d_in[19];
    const float* be_fc2 = (const float*)d_in[20];
    float* out = (float*)d_out;

    // workspace carve (256B aligned)
    size_t off = 0;
    auto carve = [&](size_t elems) -> float* {
        float* p = (float*)((char*)d_ws + off);
        off += ((elems * sizeof(float) + 255) / 256) * 256;
        return p;
    };
    float* xf      = carve((size_t)B_ * C_ * N_);
    float* y       = carve((size_t)B_ * C_ * NP_);
    float* pts     = carve((size_t)B_ * NP_ * C_);
    float* sq      = carve((size_t)B_ * NP_);
    float* dist    = carve((size_t)B_ * NP_ * NP_);
    float* density = carve((size_t)B_ * NP_);
    float* score   = carve((size_t)B_ * NP_);
    float* cent    = carve((size_t)B_ * M_ * C_);
    float* assign  = carve((size_t)B_ * N_ * MP_);
    float* colsum  = carve((size_t)B_ * MP_);
    float* aggc    = carve((size_t)B_ * C_ * MP_);
    float* hdn     = carve((size_t)B_ * C4_ * MP_);
    float* refined = carve((size_t)B_ * C_ * MP_);
    float* xcat    = carve((size_t)B_ * C2_ * N_);
    float* tmp2    = carve((size_t)B_ * C2_ * N_);
    int*   dmax    = (int*)carve(B_);
    int*   nn_idx  = (int*)carve((size_t)B_ * N_ * 5);
    if (off > ws_size) return;   // workspace too small: deterministic no-op

    // 1) fc1: xf[b,o,n] = BN(W1 @ x)      (B k-stride = N_)
    launch_gemm_bk<N_>(stream, w_fc1, 0, C_, 1, x, (long long)C_ * N_, 1, N_,
                       xf, (long long)C_ * N_, N_, C_ / 16, N_ / 16, C_ / 4, 1, 0,
                       b_fc1, g_fc1, be_fc1, nullptr, 0, 0, nullptr, 0);

    // 2) pool -> y (c-major), pts (point-major)
    {
        long long n = (long long)B_ * C_ * NP_;
        pool_kernel<<<(unsigned)((n + 255) / 256), 256, 0, stream>>>(xf, y, pts);
    }
    // 3) |pts|^2
    sq_kernel<<<(B_ * NP_ + 255) / 256, 256, 0, stream>>>(pts, sq);

    // 4) dist[b,i,j] = sqrt(max(sq_i+sq_j-2*pts.ptsT,1e-12))/sqrt(C)  (BK=NP_)
    launch_gemm_bk<NP_>(stream, pts, (long long)NP_ * C_, C_, 1,
                        y, (long long)C_ * NP_, 1, NP_,
                        dist, (long long)NP_ * NP_, NP_, NP_ / 16, NP_ / 16, C_ / 4, 2, 0,
                        nullptr, nullptr, nullptr, nullptr, 0, 0, sq, NP_);

    // 5-7) density, dmax, score
    init_dmax_kernel<<<1, 32, 0, stream>>>(dmax);
    density_kernel<<<(B_ * NP_ + 255) / 256, 256, 0, stream>>>(dist, density, dmax);
    score_kernel<<<(B_ * NP_ + 255) / 256, 256, 0, stream>>>(dist, density, dmax, score);

    // 8) top-196 centroids by rank
    select_kernel<<<(B_ * NP_ + 255) / 256, 256, 0, stream>>>(score, pts, cent);

    // 9) sim[b,n,m] = xf^T . cent^T  (stored padded to MP columns; BK=1 -> b64 B loads)
    launch_gemm_bk<1>(stream, xf, (long long)C_ * N_, 1, N_,
                      cent, (long long)M_ * C_, C_, M_,
                      assign, (long long)N_ * MP_, MP_, N_ / 16, MP_ / 16, C_ / 4, 0, 0,
                      nullptr, nullptr, nullptr, nullptr, 0, 0, nullptr, 0);

    // 10) softmax over m, 11) column sums
    {
        long long n = (long long)B_ * N_;
        softmax_kernel<<<(unsigned)((n + 255) / 256), 256, 0, stream>>>(assign);
    }
    colsum_kernel<<<(B_ * MP_ + 255) / 256, 256, 0, stream>>>(assign, colsum);

    // 12) aggc[b,c,m] = (xf @ assign) / (colsum+1e-6)   (BK=MP_)
    launch_gemm_bk<MP_>(stream, xf, (long long)C_ * N_, N_, 1,
                        assign, (long long)N_ * MP_, 1, MP_,
                        aggc, (long long)C_ * MP_, MP_, C_ / 16, MP_ / 16, N_ / 4, 3, 0,
                        nullptr, nullptr, nullptr, nullptr, 0, 0, colsum, MP_);

    // 13) ffn1 (+GELU)   (BK=MP_)
    launch_gemm_bk<MP_>(stream, w_f1, 0, C_, 1,
                        aggc, (long long)C_ * MP_, 1, MP_,
                        hdn, (long long)C4_ * MP_, MP_, C4_ / 16, MP_ / 16, C_ / 4, 1, 1,
                        b_f1, g_f1, be_f1, nullptr, 0, 0, nullptr, 0);

    // 14) ffn2 + residual(aggc) -> refined   (BK=MP_)
    launch_gemm_bk<MP_>(stream, w_f2, 0, C4_, 1,
                        hdn, (long long)C4_ * MP_, 1, MP_,
                        refined, (long long)C_ * MP_, MP_, C_ / 16, MP_ / 16, C4_ / 4, 1, 0,
                        b_f2, g_f2, be_f2, aggc, (long long)C_ * MP_, MP_, nullptr, 0);

    // 15) top-5 neighbors, 16) xcat (interleaved channels)
    {
        long long n = (long long)B_ * N_;
        topk5_kernel<<<(unsigned)((n + 255) / 256), 256, 0, stream>>>(assign, nn_idx);
        long long n2 = (long long)B_ * C_ * N_;
        xcat_kernel<<<(unsigned)((n2 + 255) / 256), 256, 0, stream>>>(xf, refined, nn_idx, xcat);
    }

    // 17) nn conv (+GELU)   (BK=N_)
    launch_gemm_bk<N_>(stream, w_nn, 0, C2_, 1,
                       xcat, (long long)C2_ * N_, 1, N_,
                       tmp2, (long long)C2_ * N_, N_, C2_ / 16, N_ / 16, C2_ / 4, 1, 1,
                       b_nn, g_nn, be_nn, nullptr, 0, 0, nullptr, 0);

    // 18) fc2 + residual(x) -> out   (BK=N_)
    launch_gemm_bk<N_>(stream, w_fc2, 0, C2_, 1,
                       tmp2, (long long)C2_ * N_, 1, N_,
                       out, (long long)C_ * N_, N_, C_ / 16, N_ / 16, C2_ / 4, 1, 0,
                       b_fc2, g_fc2, be_fc2, x, (long long)C_ * N_, N_, nullptr, 0);
}